// OnlyGATNetLeakyReLU_45818711113965
// MI455X (gfx1250) — compile-verified
//
#include <hip/hip_runtime.h>
#include <math.h>

typedef float v2f __attribute__((ext_vector_type(2)));
typedef float v8f __attribute__((ext_vector_type(8)));

#define N_NODES 8192
#define N_EDGES 262144

// ---------------------------------------------------------------- utilities
__global__ void fill_kernel(float* __restrict__ p, float v, int n) {
  int i = blockIdx.x * blockDim.x + threadIdx.x;
  if (i < n) p[i] = v;
}

__device__ __forceinline__ float atomicMaxF(float* addr, float value) {
  if (value >= 0.0f)
    return __int_as_float(atomicMax((int*)addr, __float_as_int(value)));
  else
    return __uint_as_float(atomicMin((unsigned int*)addr, __float_as_uint(value)));
}

__device__ __forceinline__ void edge_sd(const int* __restrict__ ei, int e,
                                        int& s, int& d) {
  if (e < N_EDGES) { s = ei[e]; d = ei[N_EDGES + e]; }
  else             { s = e - N_EDGES; d = s; }   // self loop
}

// -------------------------------------------------- WMMA fp32 GEMM (h = A @ W^T)
// A: [M,K] row-major, W: [N,K] row-major, C: [M,N] row-major.
// One wave (32 threads) per 16x16 output tile; V_WMMA_F32_16X16X4_F32, K step 4.
__global__ void gemm_wmma_f32(const float* __restrict__ A,
                              const float* __restrict__ W,
                              float* __restrict__ C,
                              int M, int K, int N) {
  const int lane = threadIdx.x;      // 0..31, one wave
  const int half = lane >> 4;        // selects K pair {0,1} or {2,3}
  const int l    = lane & 15;        // M (for A) / N (for B) index
  const int row0 = blockIdx.x * 16;
  const int col0 = blockIdx.y * 16;

  const int  wc  = col0 + l;
  const bool wok = wc < N;
  const int  wcc = wok ? wc : (N - 1);              // clamped, branchless guard
  const float* __restrict__ arow = A + (size_t)(row0 + l) * K + 2 * half;
  const float* __restrict__ wrow = W + (size_t)wcc * K + 2 * half;

  v8f acc = {};
  for (int k0 = 0; k0 < K; k0 += 4) {
    __builtin_prefetch(arow + k0 + 32, 0, 0);       // global_prefetch_b8
    v2f a, b;
    a.x = arow[k0];
    a.y = arow[k0 + 1];
    float b0 = wrow[k0];
    float b1 = wrow[k0 + 1];
    b.x = wok ? b0 : 0.0f;
    b.y = wok ? b1 : 0.0f;
    // 8 args: (neg_a, A, neg_b, B, c_mod, C, reuse_a, reuse_b)
    acc = __builtin_amdgcn_wmma_f32_16x16x4_f32(false, a, false, b,
                                                (short)0, acc, false, false);
  }
#pragma unroll
  for (int r = 0; r < 8; ++r) {
    int row = row0 + r + 8 * half;                  // C VGPR r -> M=r / M=r+8
    if (wok) C[(size_t)row * N + wc] = acc[r];
  }
}

// ---------------------------------------------------------------- attention
// asrc[n,h] = sum_c h[n,h,c]*att_src[h,c]; likewise adst.
__global__ void att_scores_kernel(const float* __restrict__ h,
                                  const float* __restrict__ att_s,
                                  const float* __restrict__ att_d,
                                  float* __restrict__ asrc,
                                  float* __restrict__ adst,
                                  int n, int H, int C) {
  int i = blockIdx.x * blockDim.x + threadIdx.x;
  if (i >= n * H) return;
  int node = i / H, hh = i % H;
  const float* hp = h + (size_t)node * H * C + (size_t)hh * C;
  const float* as = att_s + hh * C;
  const float* ad = att_d + hh * C;
  float s = 0.0f, d = 0.0f;
  for (int c = 0; c < C; ++c) { s = fmaf(hp[c], as[c], s); d = fmaf(hp[c], ad[c], d); }
  asrc[i] = s;
  adst[i] = d;
}

// --------------------------------------------- segment softmax over incoming edges
__global__ void edge_max_kernel(const int* __restrict__ ei,
                                const float* __restrict__ asrc,
                                const float* __restrict__ adst,
                                float* __restrict__ m, int Etot, int H) {
  int i = blockIdx.x * blockDim.x + threadIdx.x;
  if (i >= Etot * H) return;
  int e = i / H, hh = i % H;
  int s, d; edge_sd(ei, e, s, d);
  float v = asrc[s * H + hh] + adst[d * H + hh];
  v = v > 0.0f ? v : 0.2f * v;                      // leaky_relu slope 0.2
  atomicMaxF(&m[d * H + hh], v);
}

__global__ void edge_exp_kernel(const int* __restrict__ ei,
                                const float* __restrict__ asrc,
                                const float* __restrict__ adst,
                                const float* __restrict__ m,
                                float* __restrict__ denom,
                                float* __restrict__ exb, int Etot, int H) {
  int i = blockIdx.x * blockDim.x + threadIdx.x;
  if (i >= Etot * H) return;
  int e = i / H, hh = i % H;
  int s, d; edge_sd(ei, e, s, d);
  float v = asrc[s * H + hh] + adst[d * H + hh];
  v = v > 0.0f ? v : 0.2f * v;
  float ex = expf(v - m[d * H + hh]);
  exb[i] = ex;
  atomicAdd(&denom[d * H + hh], ex);
}

// out[dst, h*C+c] += (ex/denom[dst,h]) * hfeat[src, h*C+c]
__global__ void edge_agg_kernel(const int* __restrict__ ei,
                                const float* __restrict__ exb,
                                const float* __restrict__ denom,
                                const float* __restrict__ hfeat,
                                float* __restrict__ out,
                                int Etot, int H, int C) {
  int e = blockIdx.x;
  int t = threadIdx.x;
  int F = H * C;
  if (t >= F) return;
  int s, d; edge_sd(ei, e, s, d);
  int hh = t / C;
  float alpha = exb[(size_t)e * H + hh] / denom[d * H + hh];
  atomicAdd(&out[(size_t)d * F + t], alpha * hfeat[(size_t)s * F + t]);
}

__global__ void bias_act_kernel(const float* __restrict__ in,
                                const float* __restrict__ b,
                                float* __restrict__ out,
                                int n, int F, float slope) {
  int i = blockIdx.x * blockDim.x + threadIdx.x;
  if (i >= n * F) return;
  float v = in[i] + b[i % F];
  out[i] = v > 0.0f ? v : slope * v;
}

// ---------------------------------------------------------------- cdist
__global__ void sq_kernel(const float* __restrict__ z, float* __restrict__ sq, int n) {
  int i = blockIdx.x * blockDim.x + threadIdx.x;
  if (i >= n) return;
  float a = z[3 * i], b = z[3 * i + 1], c = z[3 * i + 2];
  sq[i] = a * a + b * b + c * c;
}

__global__ void cdist_kernel(const float* __restrict__ z,
                             const float* __restrict__ sq,
                             float* __restrict__ out, int n) {
  int j = blockIdx.x * blockDim.x + threadIdx.x;
  int i = blockIdx.y;
  if (j >= n) return;
  float zi0 = z[3 * i], zi1 = z[3 * i + 1], zi2 = z[3 * i + 2];
  float dot = zi0 * z[3 * j] + zi1 * z[3 * j + 1] + zi2 * z[3 * j + 2];
  float d2 = sq[i] + sq[j] - 2.0f * dot;
  d2 = fmaxf(d2, 0.0f);
  out[(size_t)i * n + j] = (d2 > 0.0f) ? sqrtf(d2) : 0.0f;
}

// ---------------------------------------------------------------- one GAT layer
static void gat_layer(const float* xin, const int* ei, const float* W,
                      const float* att_s, const float* att_d, const float* b,
                      int Kin, int H, int C, float slope,
                      float* hbuf, float* aggbuf,
                      float* asrc, float* adst, float* mbuf, float* den,
                      float* exb, hipStream_t stream) {
  const int n = N_NODES, Etot = N_EDGES + N_NODES;
  const int F = H * C;
  // 1. h = x @ W^T via WMMA
  dim3 ggrid(n / 16, (F + 15) / 16);
  gemm_wmma_f32<<<ggrid, 32, 0, stream>>>(xin, W, hbuf, n, Kin, F);
  // 2. attention scores
  int tNH = n * H;
  att_scores_kernel<<<(tNH + 255) / 256, 256, 0, stream>>>(hbuf, att_s, att_d,
                                                           asrc, adst, n, H, C);
  // 3. init segment-softmax state + accumulator
  fill_kernel<<<(tNH + 255) / 256, 256, 0, stream>>>(mbuf, -INFINITY, tNH);
  fill_kernel<<<(tNH + 255) / 256, 256, 0, stream>>>(den, 0.0f, tNH);
  int tNF = n * F;
  fill_kernel<<<(tNF + 255) / 256, 256, 0, stream>>>(aggbuf, 0.0f, tNF);
  // 4-5. segment max, then exp + segment sum
  int tEH = Etot * H;
  edge_max_kernel<<<(tEH + 255) / 256, 256, 0, stream>>>(ei, asrc, adst, mbuf, Etot, H);
  edge_exp_kernel<<<(tEH + 255) / 256, 256, 0, stream>>>(ei, asrc, adst, mbuf, den,
                                                         exb, Etot, H);
  // 6. weighted scatter-aggregate
  int bt = ((F + 31) / 32) * 32;
  edge_agg_kernel<<<Etot, bt, 0, stream>>>(ei, exb, den, hbuf, aggbuf, Etot, H, C);
  // 7. bias + inter-layer activation (slope=1 -> identity)
  bias_act_kernel<<<(tNF + 255) / 256, 256, 0, stream>>>(aggbuf, b, aggbuf, n, F, slope);
}

extern "C" void kernel_launch(void* const* d_in, const int* in_sizes, int n_in,
                              void* d_out, int out_size, void* d_ws, size_t ws_size,
                              hipStream_t stream) {
  (void)in_sizes; (void)n_in; (void)out_size; (void)ws_size;
  const float* x   = (const float*)d_in[0];
  const int*   ei  = (const int*)d_in[1];
  const float* W1  = (const float*)d_in[2];
  const float* as1 = (const float*)d_in[3];
  const float* ad1 = (const float*)d_in[4];
  const float* b1  = (const float*)d_in[5];
  const float* W2  = (const float*)d_in[6];
  const float* as2 = (const float*)d_in[7];
  const float* ad2 = (const float*)d_in[8];
  const float* b2  = (const float*)d_in[9];
  const float* W3  = (const float*)d_in[10];
  const float* as3 = (const float*)d_in[11];
  const float* ad3 = (const float*)d_in[12];
  const float* b3  = (const float*)d_in[13];
  float* out = (float*)d_out;

  const int n = N_NODES, Etot = N_EDGES + N_NODES;

  // scratch carving (all fp32), ~28 MB total
  float* p    = (float*)d_ws;
  float* h1   = p; p += (size_t)n * 256;
  float* a1   = p; p += (size_t)n * 256;
  float* h2   = p; p += (size_t)n * 128;
  float* a2   = p; p += (size_t)n * 128;
  float* h3   = p; p += (size_t)n * 3;
  float* zb   = p; p += (size_t)n * 3;
  float* asrc = p; p += (size_t)n * 2;
  float* adst = p; p += (size_t)n * 2;
  float* mbuf = p; p += (size_t)n * 2;
  float* den  = p; p += (size_t)n * 2;
  float* exb  = p; p += (size_t)Etot * 2;
  float* sq   = p; p += n;

  // layer 1: in=512 -> heads=2, C=128, concat -> 256, leaky 0.01
  gat_layer(x,  ei, W1, as1, ad1, b1, 512, 2, 128, 0.01f,
            h1, a1, asrc, adst, mbuf, den, exb, stream);
  // layer 2: in=256 -> heads=2, C=64, concat -> 128, leaky 0.01
  gat_layer(a1, ei, W2, as2, ad2, b2, 256, 2, 64, 0.01f,
            h2, a2, asrc, adst, mbuf, den, exb, stream);
  // layer 3: in=128 -> heads=1, C=3, mean(1 head)=identity, no activation
  gat_layer(a2, ei, W3, as3, ad3, b3, 128, 1, 3, 1.0f,
            h3, zb, asrc, adst, mbuf, den, exb, stream);

  // pairwise euclidean distances (reference formula, incl. zero-diagonal guard)
  sq_kernel<<<(n + 255) / 256, 256, 0, stream>>>(zb, sq, n);
  cdist_kernel<<<dim3(n / 256, n), 256, 0, stream>>>(zb, sq, out, n);
}